// _LightGCNModel_31628139168302
// MI455X (gfx1250) — compile-verified
//
#include <hip/hip_runtime.h>
#include <hip/hip_bf16.h>

#define EMB 64

typedef __attribute__((ext_vector_type(2))) float v2f;
typedef __attribute__((ext_vector_type(8))) float v8f;

// ---------------------------------------------------------------------------
// Init: cur = concat(user_emb, item_emb); acc = cur; nxt = 0
// ---------------------------------------------------------------------------
__global__ void lg_init_kernel(const float* __restrict__ ue,
                               const float* __restrict__ ie,
                               float* __restrict__ cur,
                               float* __restrict__ acc,
                               float* __restrict__ nxt,
                               int nu_elems, int total) {
    int i = blockIdx.x * blockDim.x + threadIdx.x;
    if (i >= total) return;
    float v = (i < nu_elems) ? ue[i] : ie[i - nu_elems];
    cur[i] = v;
    acc[i] = v;
    nxt[i] = 0.0f;
}

// ---------------------------------------------------------------------------
// SpMM: one wave per edge. Lane d covers dims {2d, 2d+1} as a float2 so the
// 256B source row gather and the atomic scatter are fully coalesced. The
// embedding table (35.8 MB) is L2-resident on MI455X (192 MB L2), so the
// f32 atomics resolve at L2 bandwidth.
// ---------------------------------------------------------------------------
__global__ void lg_spmm_kernel(const float* __restrict__ src,
                               float* __restrict__ dst,
                               const float* __restrict__ vals,
                               const int* __restrict__ rows,
                               const int* __restrict__ cols,
                               int nedges) {
    int lane = threadIdx.x & 31;
    int e = blockIdx.x * (blockDim.x >> 5) + (threadIdx.x >> 5);
    if (e >= nedges) return;
    float w = vals[e];
    int r = rows[e];
    int c = cols[e];
    const float2* s = (const float2*)(src + (size_t)c * EMB);
    float2 x = s[lane];
    float* d = dst + (size_t)r * EMB + 2 * lane;
    __hip_atomic_fetch_add(d,     w * x.x, __ATOMIC_RELAXED, __HIP_MEMORY_SCOPE_AGENT);
    __hip_atomic_fetch_add(d + 1, w * x.y, __ATOMIC_RELAXED, __HIP_MEMORY_SCOPE_AGENT);
}

// ---------------------------------------------------------------------------
// Fused layer epilogue: acc += newemb ; zerobuf = 0 (prepares next SpMM dst)
// ---------------------------------------------------------------------------
__global__ void lg_accum_kernel(float* __restrict__ acc,
                                const float* __restrict__ newemb,
                                float* __restrict__ zerobuf,
                                int total) {
    int i = blockIdx.x * blockDim.x + threadIdx.x;
    if (i >= total) return;
    acc[i] += newemb[i];
    zerobuf[i] = 0.0f;
}

// ---------------------------------------------------------------------------
// Final batched dot via V_WMMA_F32_16X16X4_F32 (full fp32 precision).
// One wave handles 16 batch elements: D = U(16x64) * I^T(64x16) built from
// 16 chained K=4 WMMA steps; gamma = diag(D) / 16 (the /4 on each factor).
//
// A 16x4 f32 layout: lane m=lane%16 holds row M=m; VGPR v + lane-half h give
// K = 2h + v. B 4x16 mirrored: lane n=lane%16 is column N. C/D: VGPR r holds
// M=r (lanes 0-15) / M=r+8 (lanes 16-31), N = lane%16.
// ---------------------------------------------------------------------------
__global__ void lg_dot_wmma_kernel(const float* __restrict__ acc,
                                   const int* __restrict__ users,
                                   const int* __restrict__ items,
                                   float* __restrict__ out,
                                   int n_users, int batch) {
    __shared__ float smem[8 * 256]; // 8 waves/block * (8 VGPR rows * 32 lanes)
    int lane = threadIdx.x & 31;
    int wv   = threadIdx.x >> 5;
    int tile = blockIdx.x * (blockDim.x >> 5) + wv; // 16 batch elems per tile
    if (tile * 16 >= batch) return;                 // launch is exact: never taken

    int m = lane & 15;      // row of this lane's A / col of this lane's B
    int h = lane >> 4;      // lane-half selects K pair within each K=4 block

    int b = tile * 16 + m;
    const float* urow = acc + (size_t)users[b] * EMB;
    const float* irow = acc + (size_t)(n_users + items[b]) * EMB;

    v8f c = {};
#pragma unroll
    for (int kk = 0; kk < 16; ++kk) {
        int k = kk * 4 + h * 2;
        v2f a;  a[0] = urow[k];  a[1] = urow[k + 1];
        v2f bm; bm[0] = irow[k]; bm[1] = irow[k + 1];
        // 8 args: (neg_a, A, neg_b, B, c_mod, C, reuse_a, reuse_b)
        c = __builtin_amdgcn_wmma_f32_16x16x4_f32(
                false, a, false, bm, (short)0, c, false, false);
    }

    // Spill the 16x16 tile to LDS and read back the diagonal. Same-wave DS
    // ops execute in order, so no cross-wave barrier is required.
    float* ws = smem + wv * 256;
#pragma unroll
    for (int r = 0; r < 8; ++r) ws[r * 32 + lane] = c[r];
    asm volatile("" ::: "memory");

    if (lane < 16) {
        int d = lane;
        float g = (d < 8) ? ws[d * 32 + d]               // M=r=d, N=lane=d
                          : ws[(d - 8) * 32 + 16 + d];   // M=r+8=d, N=lane-16=d
        out[tile * 16 + d] = g * 0.0625f;                // (acc/4)·(acc/4)
    }
}

// ---------------------------------------------------------------------------
// Host launcher. Workspace layout: cur | nxt | acc, each total*4 bytes
// (total = 140000*64 -> 107.5 MB of d_ws used).
// ---------------------------------------------------------------------------
extern "C" void kernel_launch(void* const* d_in, const int* in_sizes, int n_in,
                              void* d_out, int out_size, void* d_ws, size_t ws_size,
                              hipStream_t stream) {
    const float* user_emb = (const float*)d_in[0];
    const float* item_emb = (const float*)d_in[1];
    const float* gvals    = (const float*)d_in[2];
    const int*   grow     = (const int*)d_in[3];
    const int*   gcol     = (const int*)d_in[4];
    const int*   users    = (const int*)d_in[5];
    const int*   items    = (const int*)d_in[6];

    const int nu_elems = in_sizes[0];           // N_USERS * 64
    const int ni_elems = in_sizes[1];           // N_ITEMS * 64
    const int nedges   = in_sizes[2];           // 2 * NNZ
    const int batch    = in_sizes[5];
    const int n_users  = nu_elems / EMB;
    const int total    = nu_elems + ni_elems;   // node embedding elements

    float* cur = (float*)d_ws;
    float* nxt = cur + total;
    float* acc = nxt + total;

    const int ethreads = 256;
    const int eblocks  = (total + ethreads - 1) / ethreads;

    lg_init_kernel<<<eblocks, ethreads, 0, stream>>>(
        user_emb, item_emb, cur, acc, nxt, nu_elems, total);

    const int edges_per_block = 256 / 32;       // 8 waves -> 8 edges per block
    const int sblocks = (nedges + edges_per_block - 1) / edges_per_block;

    for (int layer = 0; layer < 3; ++layer) {
        lg_spmm_kernel<<<sblocks, 256, 0, stream>>>(cur, nxt, gvals, grow, gcol, nedges);
        lg_accum_kernel<<<eblocks, ethreads, 0, stream>>>(acc, nxt, cur, total);
        float* t = cur; cur = nxt; nxt = t;     // cur now holds this layer's emb
    }

    const int tiles  = batch / 16;              // 1024
    const int dblk   = tiles / 8;               // 8 waves per block -> exact launch
    lg_dot_wmma_kernel<<<dblk, 256, 0, stream>>>(acc, users, items,
                                                 (float*)d_out, n_users, batch);
}